// GATDecoder_86870008529058
// MI455X (gfx1250) — compile-verified
//
#include <hip/hip_runtime.h>
#include <hip/hip_bf16.h>
#include <math.h>

// ---------------------------------------------------------------------------
// Hyperbolic GAT decoder for MI455X (gfx1250, wave32, WMMA).
// Node GEMMs use V_WMMA_F32_16X16X4_F32 (exact f32 path; problem is
// HBM-bandwidth-bound so low-precision WMMA would gain nothing).
// ---------------------------------------------------------------------------

typedef __attribute__((ext_vector_type(2))) float v2f;
typedef __attribute__((ext_vector_type(8))) float v8f;

#define IN_FEAT   256
#define OUTC      16     // HEADS * N_CLASSES
#define NEG_SLOPE 0.2f
#define EPSF      1e-15f

// order-preserving float<->uint map for atomicMax on floats
__device__ __forceinline__ unsigned fmap(float f) {
    unsigned u = __float_as_uint(f);
    return (u & 0x80000000u) ? ~u : (u | 0x80000000u);
}
__device__ __forceinline__ float funmap(unsigned u) {
    unsigned v = (u & 0x80000000u) ? (u & 0x7FFFFFFFu) : ~u;
    return __uint_as_float(v);
}

// ---------------------------------------------------------------------------
// Kernel 1: per-16-node tile, one wave each.
//   cf = X_tile @ W       (f32 WMMA, K stepped by 4, 64 iterations)
//   cr = X_tile @ W_res   (shares the A fragment)
//   norm^2 of each row accumulated during A loads, completed via shfl_xor(16);
//   logmap0 scale applied to the 16x16 D tiles; bias folded into rst.
// ---------------------------------------------------------------------------
__global__ __launch_bounds__(256) void k_node(
    const float* __restrict__ x, const float* __restrict__ W,
    const float* __restrict__ Wr, const float* __restrict__ bias,
    float* __restrict__ feat, float* __restrict__ rst, int N, int nTiles)
{
    __shared__ float sW [IN_FEAT * OUTC];
    __shared__ float sWr[IN_FEAT * OUTC];
    int tid = threadIdx.x;
    for (int i = tid; i < IN_FEAT * OUTC; i += 256) {
        sW[i]  = W[i];
        sWr[i] = Wr[i];
    }
    __syncthreads();

    int wave = tid >> 5;
    int lane = tid & 31;
    int t = blockIdx.x * 8 + wave;
    if (t >= nTiles) t = nTiles - 1;      // duplicate last tile: benign identical writes

    int half = lane >> 4;                 // 0: K=k0..k0+1   1: K=k0+2..k0+3
    int l15  = lane & 15;                 // row (A) / column (B,D)
    int rowg = t * 16 + l15;
    if (rowg >= N) rowg = N - 1;
    const float* xrow = x + (size_t)rowg * IN_FEAT;

    v8f cf = {0.f,0.f,0.f,0.f,0.f,0.f,0.f,0.f};
    v8f cr = {0.f,0.f,0.f,0.f,0.f,0.f,0.f,0.f};
    float nsq = 0.f;

    for (int k0 = 0; k0 < IN_FEAT; k0 += 4) {
        int ka = k0 + half * 2;
        // A fragment (16x4 f32): lane row = l15, two consecutive K values
        float a0 = xrow[ka], a1 = xrow[ka + 1];
        nsq = fmaf(a0, a0, nsq);
        nsq = fmaf(a1, a1, nsq);
        v2f a; a[0] = a0; a[1] = a1;
        // B fragments (4x16 f32): VGPR0 = row ka, VGPR1 = row ka+1, column l15
        v2f b;  b[0]  = sW [ka * OUTC + l15];  b[1]  = sW [(ka + 1) * OUTC + l15];
        v2f br; br[0] = sWr[ka * OUTC + l15];  br[1] = sWr[(ka + 1) * OUTC + l15];
        cf = __builtin_amdgcn_wmma_f32_16x16x4_f32(false, a, false, b,  (short)0, cf, false, false);
        cr = __builtin_amdgcn_wmma_f32_16x16x4_f32(false, a, false, br, (short)0, cr, false, false);
    }

    // complete per-row norm^2: lane l and l+16 hold disjoint K partials of row l
    nsq += __shfl_xor(nsq, 16, 32);
    float nrm = fmaxf(sqrtf(nsq), EPSF);
    float z   = fminf(nrm, 1.0f - 1e-7f);           // artanh clip (norm >= 0)
    float sc  = 0.5f * (log1pf(z) - log1pf(-z)) / nrm;
    float bv  = bias[l15];

    float* fp = feat + (size_t)t * 16 * OUTC;
    float* rp = rst  + (size_t)t * 16 * OUTC;
    #pragma unroll
    for (int r = 0; r < 8; ++r) {
        int row = r + half * 8;                      // D layout: M=r (lanes 0-15), M=r+8 (16-31)
        float s_r = __shfl(sc, row, 32);             // scale of row `row` lives in lane `row`
        if (t * 16 + row < N) {
            fp[row * OUTC + l15] = cf[r] * s_r;
            rp[row * OUTC + l15] = cr[r] * s_r + bv; // residual + bias, pre-aggregation
        }
    }
}

// Kernel 2: el/er per (node, head)
__global__ void k_elr(const float* __restrict__ feat, const float* __restrict__ al,
                      const float* __restrict__ ar, float* __restrict__ el,
                      float* __restrict__ er, int N)
{
    int n = blockIdx.x * blockDim.x + threadIdx.x;
    if (n >= N) return;
    const float* f = feat + (size_t)n * OUTC;
    #pragma unroll
    for (int h = 0; h < 2; ++h) {
        float sl = 0.f, sr = 0.f;
        #pragma unroll
        for (int c = 0; c < 8; ++c) {
            float v = f[h * 8 + c];
            sl = fmaf(v, al[h * 8 + c], sl);
            sr = fmaf(v, ar[h * 8 + c], sr);
        }
        el[n * 2 + h] = sl;
        er[n * 2 + h] = sr;
    }
}

// Kernel 3: init segment-max (mapped -inf) and denominators
__global__ void k_init(unsigned* __restrict__ mmax, float* __restrict__ denom, int n2)
{
    int i = blockIdx.x * blockDim.x + threadIdx.x;
    if (i < n2) { mmax[i] = 0x007FFFFFu; denom[i] = 0.f; }   // fmap(-inf)
}

// Kernel 4: segment max of leaky(el[src]+er[dst]) per destination
__global__ void k_edge_max(const int* __restrict__ ei, const float* __restrict__ el,
                           const float* __restrict__ er, unsigned* __restrict__ mmax, int E)
{
    int e = blockIdx.x * blockDim.x + threadIdx.x;
    if (e >= E) return;
    int s = ei[e], d = ei[E + e];
    #pragma unroll
    for (int h = 0; h < 2; ++h) {
        float v = el[s * 2 + h] + er[d * 2 + h];
        v = (v >= 0.f) ? v : NEG_SLOPE * v;
        atomicMax(&mmax[d * 2 + h], fmap(v));
    }
}

// Kernel 5: unmap + replace non-finite (isolated nodes) with 0
__global__ void k_max_fix(const unsigned* __restrict__ mmax, float* __restrict__ mfix, int n2)
{
    int i = blockIdx.x * blockDim.x + threadIdx.x;
    if (i >= n2) return;
    float m = funmap(mmax[i]);
    mfix[i] = (fabsf(m) <= 3.402823466e38f) ? m : 0.f;
}

// Kernel 6: softmax denominators
__global__ void k_edge_sum(const int* __restrict__ ei, const float* __restrict__ el,
                           const float* __restrict__ er, const float* __restrict__ mfix,
                           float* __restrict__ denom, int E)
{
    int e = blockIdx.x * blockDim.x + threadIdx.x;
    if (e >= E) return;
    int s = ei[e], d = ei[E + e];
    #pragma unroll
    for (int h = 0; h < 2; ++h) {
        float v = el[s * 2 + h] + er[d * 2 + h];
        v = (v >= 0.f) ? v : NEG_SLOPE * v;
        atomicAdd(&denom[d * 2 + h], __expf(v - mfix[d * 2 + h]));
    }
}

// Kernel 7: alpha-weighted scatter-add of source features into rst
__global__ void k_edge_agg(const int* __restrict__ ei, const float* __restrict__ el,
                           const float* __restrict__ er, const float* __restrict__ mfix,
                           const float* __restrict__ denom, const float* __restrict__ feat,
                           float* __restrict__ rst, int E)
{
    int e = blockIdx.x * blockDim.x + threadIdx.x;
    if (e >= E) return;
    int s = ei[e], d = ei[E + e];
    float alpha[2];
    #pragma unroll
    for (int h = 0; h < 2; ++h) {
        float v = el[s * 2 + h] + er[d * 2 + h];
        v = (v >= 0.f) ? v : NEG_SLOPE * v;
        float ex = __expf(v - mfix[d * 2 + h]);
        alpha[h] = ex / fmaxf(denom[d * 2 + h], EPSF);
    }
    const float* fs = feat + (size_t)s * OUTC;
    float* rd = rst + (size_t)d * OUTC;
    #pragma unroll
    for (int j = 0; j < OUTC; ++j)
        atomicAdd(&rd[j], alpha[j >> 3] * fs[j]);
}

// Kernel 8: head mean -> [N, 8]
__global__ void k_out(const float* __restrict__ rst, float* __restrict__ out, int N)
{
    int i = blockIdx.x * blockDim.x + threadIdx.x;
    if (i >= N * 8) return;
    int n = i >> 3, c = i & 7;
    out[i] = 0.5f * (rst[(size_t)n * OUTC + c] + rst[(size_t)n * OUTC + 8 + c]);
}

// ---------------------------------------------------------------------------
extern "C" void kernel_launch(void* const* d_in, const int* in_sizes, int n_in,
                              void* d_out, int out_size, void* d_ws, size_t ws_size,
                              hipStream_t stream)
{
    const float* x    = (const float*)d_in[0];
    const int*   ei   = (const int*)  d_in[1];   // edge_index, [2, E]
    const float* W    = (const float*)d_in[2];
    const float* al   = (const float*)d_in[3];
    const float* ar   = (const float*)d_in[4];
    const float* bias = (const float*)d_in[5];
    const float* Wr   = (const float*)d_in[6];
    float* out = (float*)d_out;

    int N = in_sizes[0] / IN_FEAT;
    int E = in_sizes[1] / 2;

    // workspace carve-up (~16.8 MB for N=1e5)
    char* ws = (char*)d_ws;
    float*    feat  = (float*)ws;    ws += (size_t)N * OUTC * sizeof(float);
    float*    rst   = (float*)ws;    ws += (size_t)N * OUTC * sizeof(float);
    float*    el    = (float*)ws;    ws += (size_t)N * 2 * sizeof(float);
    float*    er    = (float*)ws;    ws += (size_t)N * 2 * sizeof(float);
    float*    mfix  = (float*)ws;    ws += (size_t)N * 2 * sizeof(float);
    float*    denom = (float*)ws;    ws += (size_t)N * 2 * sizeof(float);
    unsigned* mmax  = (unsigned*)ws; ws += (size_t)N * 2 * sizeof(unsigned);

    int nTiles = (N + 15) / 16;
    dim3 blk(256);

    k_node    <<<(nTiles + 7) / 8,      blk, 0, stream>>>(x, W, Wr, bias, feat, rst, N, nTiles);
    k_init    <<<(2 * N + 255) / 256,   blk, 0, stream>>>(mmax, denom, 2 * N);
    k_elr     <<<(N + 255) / 256,       blk, 0, stream>>>(feat, al, ar, el, er, N);
    k_edge_max<<<(E + 255) / 256,       blk, 0, stream>>>(ei, el, er, mmax, E);
    k_max_fix <<<(2 * N + 255) / 256,   blk, 0, stream>>>(mmax, mfix, 2 * N);
    k_edge_sum<<<(E + 255) / 256,       blk, 0, stream>>>(ei, el, er, mfix, denom, E);
    k_edge_agg<<<(E + 255) / 256,       blk, 0, stream>>>(ei, el, er, mfix, denom, feat, rst, E);
    k_out     <<<(N * 8 + 255) / 256,   blk, 0, stream>>>(rst, out, N);
}